// Shuffling_45732811768395
// MI455X (gfx1250) — compile-verified
//
#include <hip/hip_runtime.h>
#include <stdint.h>

// Reference: z is (n=100000, d=512) float32; out[r, c] = z[perm_c(r), c] with an
// independent permutation per column. d is fixed at 512 by the reference.
#define D_COLS 512u

// Native Clang vector type: accepted by __builtin_nontemporal_store and lowers
// to one global_store_b128.
typedef float v4f __attribute__((ext_vector_type(4)));

// ---- cheap integer mixer (murmur3-style finalizer fragment) -----------------
__device__ __forceinline__ uint32_t rmix(uint32_t x, uint32_t k) {
    x = (x ^ k) * 0x9E3779B1u;
    x ^= x >> 15;
    x *= 0x85EBCA77u;
    x ^= x >> 13;
    return x;
}

// ---- 4-round unbalanced alternating Feistel bijection on [0, 2^17) ----------
// Split 17 bits as (L:9, R:8); each round: L ^= F(R) masked to L's width, swap.
// Widths alternate 9,8,9,8 -> after 4 rounds layout is (L:9, R:8) again.
// Every round is invertible => this is a permutation of [0, 131072).
__device__ __forceinline__ uint32_t feistel17(uint32_t x, uint32_t key) {
    uint32_t L = x >> 8;      // 9 bits
    uint32_t R = x & 0xFFu;   // 8 bits
    uint32_t t;
    t = L ^ (rmix(R, key ^ 0xA511E9B3u) & 0x1FFu); L = R; R = t; // L:8 R:9
    t = L ^ (rmix(R, key ^ 0x63D83595u) & 0x0FFu); L = R; R = t; // L:9 R:8
    t = L ^ (rmix(R, key ^ 0x2545F491u) & 0x1FFu); L = R; R = t; // L:8 R:9
    t = L ^ (rmix(R, key ^ 0x9E3779B9u) & 0x0FFu); L = R; R = t; // L:9 R:8
    return (L << 8) | R;      // 17 bits
}

// Cycle-walk down to [0, n). n = 100000, domain = 131072 -> E[iters] ~ 1.31.
// Feistel is a bijection on the full domain, so walking stays a bijection on [0,n).
__device__ __forceinline__ uint32_t perm_row(uint32_t r, uint32_t key, uint32_t n) {
    uint32_t p = feistel17(r, key);
    while (p >= n) p = feistel17(p, key);
    return p;
}

// One thread handles 4 consecutive columns of one output row:
//  - 4 independent scattered gathers in flight (ILP; absorbed by 192MB L2)
//  - one coalesced 16B non-temporal store (output is write-once, keep it out of L2
//    so the 204.8MB gather working set stays ~94% L2-resident).
__global__ void __launch_bounds__(256, 8)
shuffle_cols_kernel(const float* __restrict__ z, float* __restrict__ out,
                    uint32_t n, uint32_t nquads) {
    uint32_t q = blockIdx.x * blockDim.x + threadIdx.x;
    if (q >= nquads) return;

    uint32_t r  = q >> 7;             // 512 cols / 4 = 128 quads per row
    uint32_t c0 = (q & 127u) << 2;    // starting column of this quad

    // Independent key per column (column id -> Feistel round-key seed).
    uint32_t k0 = rmix(c0 + 0u, 0x7F4A7C15u);
    uint32_t k1 = rmix(c0 + 1u, 0x7F4A7C15u);
    uint32_t k2 = rmix(c0 + 2u, 0x7F4A7C15u);
    uint32_t k3 = rmix(c0 + 3u, 0x7F4A7C15u);

    uint32_t p0 = perm_row(r, k0, n);
    uint32_t p1 = perm_row(r, k1, n);
    uint32_t p2 = perm_row(r, k2, n);
    uint32_t p3 = perm_row(r, k3, n);

    v4f v;
    v.x = z[(size_t)p0 * D_COLS + c0 + 0u];
    v.y = z[(size_t)p1 * D_COLS + c0 + 1u];
    v.z = z[(size_t)p2 * D_COLS + c0 + 2u];
    v.w = z[(size_t)p3 * D_COLS + c0 + 3u];

    v4f* dst = (v4f*)(out + (size_t)r * D_COLS + c0);
    __builtin_nontemporal_store(v, dst);   // global_store_b128 with NT hint
}

extern "C" void kernel_launch(void* const* d_in, const int* in_sizes, int n_in,
                              void* d_out, int out_size, void* d_ws, size_t ws_size,
                              hipStream_t stream) {
    const float* z = (const float*)d_in[0];
    float* out = (float*)d_out;

    uint32_t total  = (uint32_t)in_sizes[0];   // n * 512 = 51,200,000 elements
    uint32_t n      = total / D_COLS;          // 100,000 rows (< 2^17 Feistel domain)
    uint32_t nquads = total / 4u;              // 12.8M threads, 1 float4 each

    dim3 block(256);                            // 8 wave32 per block
    dim3 grid((nquads + block.x - 1u) / block.x);
    shuffle_cols_kernel<<<grid, block, 0, stream>>>(z, out, n, nquads);
}